// rqsplines_74972949119217
// MI455X (gfx1250) — compile-verified
//
#include <hip/hip_runtime.h>
#include <cstdint>

#define TAILB 20.0f
#define NB    16
#define NG    8
#define NI    64

// LDS layout (float offsets)
#define RAWW 0        // [64][16] raw W logits      (1024)
#define RAWH 1024     // [64][16] raw H logits      (1024)
#define RAWD 2048     // [64][15] raw D logits      ( 960)
#define CUMW 3008     // [17][64] knot x            (1088)
#define CUMH 4096     // [17][64] knot y            (1088)
#define DERV 5184     // [17][64] derivatives       (1088)
#define HGT  6272     // [16][64] heights           (1024)
#define DLT  7296     // [16][64] slopes h/w        (1024)
#define INVW 8320     // [16][64] 1/width           (1024)
#define SMEMF 9344

typedef int v4i __attribute__((vector_size(16)));
typedef __attribute__((address_space(1))) v4i gv4i;
typedef __attribute__((address_space(3))) v4i lv4i;

__device__ __forceinline__ void async_ld_b128(const void* gsrc, void* lds) {
#if __has_builtin(__builtin_amdgcn_global_load_async_to_lds_b128)
  __builtin_amdgcn_global_load_async_to_lds_b128(
      (gv4i*)(void*)gsrc, (lv4i*)lds, 0, 0);
#else
  unsigned int l = (unsigned int)(uintptr_t)lds;
  asm volatile("global_load_async_to_lds_b128 %0, %1, off"
               :: "v"(l), "v"(gsrc) : "memory");
#endif
}

__device__ __forceinline__ void wait_async0() {
#if __has_builtin(__builtin_amdgcn_s_wait_asynccnt)
  __builtin_amdgcn_s_wait_asynccnt(0);
#else
  asm volatile("s_wait_asynccnt 0" ::: "memory");
#endif
}

__global__ __launch_bounds__(256) void rqs_fwd_kernel(
    const float* __restrict__ X, const float* __restrict__ Wl,
    const float* __restrict__ Hl, const float* __restrict__ Dl,
    float* __restrict__ Oz, float* __restrict__ Old, int batch)
{
  __shared__ __align__(16) float sm[SMEMF];
  const int tid = threadIdx.x;
  const int g   = blockIdx.y;

  // ---- Stage raw per-g parameter slice into LDS via async DMA (ASYNCcnt) ----
  const float* Wg = Wl + g * (NI * NB);        // 1024 floats, 16B aligned
  const float* Hg = Hl + g * (NI * NB);        // 1024 floats
  const float* Dg = Dl + g * (NI * (NB - 1));  // 960 floats, 3840B (16B aligned)
  for (int c = tid; c < 752; c += 256) {       // 752 x 16B chunks
    const char* src; int ldst;
    if (c < 256)      { src = (const char*)Wg + (c << 4);         ldst = (RAWW * 4) + (c << 4); }
    else if (c < 512) { src = (const char*)Hg + ((c - 256) << 4); ldst = (RAWH * 4) + ((c - 256) << 4); }
    else              { src = (const char*)Dg + ((c - 512) << 4); ldst = (RAWD * 4) + ((c - 512) << 4); }
    async_ld_b128(src, (char*)sm + ldst);
  }
  wait_async0();
  __syncthreads();

  // ---- Build derived tables: one thread per input column i ----
  if (tid < NI) {
    const int i = tid;
    float lw[NB], lh[NB];
    float mw = -3.4e38f, mh = -3.4e38f;
    for (int k = 0; k < NB; ++k) {
      lw[k] = sm[RAWW + i * NB + k]; mw = fmaxf(mw, lw[k]);
      lh[k] = sm[RAWH + i * NB + k]; mh = fmaxf(mh, lh[k]);
    }
    float sw = 0.f, sh = 0.f;
    for (int k = 0; k < NB; ++k) {
      lw[k] = expf(lw[k] - mw); sw += lw[k];
      lh[k] = expf(lh[k] - mh); sh += lh[k];
    }
    const float fw = (2.0f * TAILB) / sw, fh = (2.0f * TAILB) / sh;
    float accw = -TAILB, acch = -TAILB;
    sm[CUMW + 0 * 64 + i] = -TAILB;
    sm[CUMH + 0 * 64 + i] = -TAILB;
    for (int k = 0; k < NB; ++k) {
      const float wk = lw[k] * fw, hk = lh[k] * fh;
      const float iw = 1.0f / wk;
      sm[INVW + k * 64 + i] = iw;
      sm[HGT  + k * 64 + i] = hk;
      sm[DLT  + k * 64 + i] = hk * iw;
      accw += wk; acch += hk;
      sm[CUMW + (k + 1) * 64 + i] = accw;
      sm[CUMH + (k + 1) * 64 + i] = acch;
    }
    sm[DERV + 0  * 64 + i] = 1.0f;
    sm[DERV + 16 * 64 + i] = 1.0f;
    for (int k = 1; k < NB; ++k) {
      const float v = sm[RAWD + i * (NB - 1) + (k - 1)];
      // jax.nn.softplus == max(v,0) + log1p(exp(-|v|))
      sm[DERV + k * 64 + i] = fmaxf(v, 0.0f) + log1pf(expf(-fabsf(v)));
    }
  }
  __syncthreads();

  // ---- Streamed elementwise spline: 64 rows per block, 16 passes x 4 rows ----
  const int i = tid & 63;
  const int r = tid >> 6;
  const int b0 = blockIdx.x * 64;
  #pragma unroll 2
  for (int p = 0; p < 16; ++p) {
    const int b = b0 + p * 4 + r;
    if (b >= batch) break;
    const int off = (b * NG + g) * NI + i;
    if (p < 15) __builtin_prefetch(&X[off + 4 * NG * NI], 0, 0);

    const float xv = __builtin_nontemporal_load(&X[off]);
    const float xc = fminf(fmaxf(xv, -TAILB), TAILB);

    // branchless binary search over monotonic knots (4 LDS loads)
    int idx = (xc >= sm[CUMW + 8 * 64 + i]) ? 8 : 0;
    idx += (xc >= sm[CUMW + (idx + 4) * 64 + i]) ? 4 : 0;
    idx += (xc >= sm[CUMW + (idx + 2) * 64 + i]) ? 2 : 0;
    idx += (xc >= sm[CUMW + (idx + 1) * 64 + i]) ? 1 : 0;

    const int kb = idx * 64 + i;               // stride 64 -> bank-conflict-free
    const float in_cw = sm[CUMW + kb];
    const float iw    = sm[INVW + kb];
    const float in_ch = sm[CUMH + kb];
    const float hh    = sm[HGT + kb];
    const float dl    = sm[DLT + kb];
    const float d0    = sm[DERV + kb];
    const float d1    = sm[DERV + kb + 64];

    const float th    = (xc - in_cw) * iw;
    const float omt   = 1.0f - th;
    const float tt    = th * omt;
    const float th2   = th * th;
    const float denom = dl + (d0 + d1 - 2.0f * dl) * tt;
    const float rd    = 1.0f / denom;
    const float z_in  = in_ch + hh * (dl * th2 + d0 * tt) * rd;
    const float dnum  = dl * dl * (d1 * th2 + 2.0f * dl * tt + d0 * omt * omt);
    const float ld_in = logf(dnum * rd * rd);

    const bool inside = (xv >= -TAILB) && (xv <= TAILB);
    __builtin_nontemporal_store(inside ? z_in  : xv,   &Oz[off]);
    __builtin_nontemporal_store(inside ? ld_in : 0.0f, &Old[off]);
  }
}

extern "C" void kernel_launch(void* const* d_in, const int* in_sizes, int n_in,
                              void* d_out, int out_size, void* d_ws, size_t ws_size,
                              hipStream_t stream) {
  (void)n_in; (void)d_ws; (void)ws_size; (void)out_size;
  const float* X  = (const float*)d_in[0];
  const float* Wl = (const float*)d_in[1];
  const float* Hl = (const float*)d_in[2];
  const float* Dl = (const float*)d_in[3];
  const int batch = in_sizes[0] / (NG * NI);
  const int N     = batch * NG * NI;
  float* Oz  = (float*)d_out;
  float* Old = Oz + N;
  dim3 grid((batch + 63) / 64, NG);
  rqs_fwd_kernel<<<grid, 256, 0, stream>>>(X, Wl, Hl, Dl, Oz, Old, batch);
}